// AdaptiveEmbedding_16836271800789
// MI455X (gfx1250) — compile-verified
//
#include <hip/hip_runtime.h>
#include <hip/hip_bf16.h>
#include <math.h>

typedef float v2f __attribute__((ext_vector_type(2)));
typedef float v8f __attribute__((ext_vector_type(8)));

#define NB   128      // batch (both Bi and Bc)
#define NR   36       // regions
#define NT   64       // max words
#define ND   1024     // latent dim
#define NH   128      // hidden dim
#define NSAMP 4608.0f // NB*NR

// ---------------------------------------------------------------------------
// 0) zero the atomic accumulators (chansum + chansumsq = 2048 floats)
__global__ void k_zero(float* __restrict__ p, int n) {
    int idx = blockIdx.x * blockDim.x + threadIdx.x;
    if (idx < n) p[idx] = 0.0f;
}

// ---------------------------------------------------------------------------
// 1) masked mean-pool of caption words -> cap_repr (NB,ND); rnorm[c]=1/||cap_repr[c]||
__global__ void k_cap_pool(const float* __restrict__ cap, const int* __restrict__ lens,
                           float* __restrict__ cap_repr, float* __restrict__ rnorm) {
    __shared__ float red[256];
    const int c   = blockIdx.x;
    const int tid = threadIdx.x;
    const int len = lens[c];
    const float inv = 1.0f / (float)len;
    float sq = 0.0f;
    for (int j = 0; j < 4; ++j) {
        const int d = j * 256 + tid;
        const float* p = cap + ((size_t)c * NT) * ND + d;
        float s = 0.0f;
        for (int t = 0; t < len; ++t) s += p[(size_t)t * ND];
        s *= inv;
        cap_repr[c * ND + d] = s;
        sq += s * s;
    }
    red[tid] = sq;
    __syncthreads();
    for (int s = 128; s > 0; s >>= 1) {
        if (tid < s) red[tid] += red[tid + s];
        __syncthreads();
    }
    if (tid == 0) rnorm[c] = rsqrtf(red[0]);
}

// ---------------------------------------------------------------------------
// 2) one pass over img_embed: per-(b,d) region sum + per-channel BN stats
__global__ void k_img_stats(const float* __restrict__ img, float* __restrict__ rowsum,
                            float* __restrict__ chansum, float* __restrict__ chansumsq) {
    const int b   = blockIdx.x;
    const int tid = threadIdx.x;
    for (int j = 0; j < 4; ++j) {
        const int d = j * 256 + tid;
        const float* p = img + ((size_t)b * NR) * ND + d;
        float s = 0.0f, s2 = 0.0f;
        for (int r = 0; r < NR; ++r) {
            float x = p[(size_t)r * ND];
            s += x; s2 += x * x;
        }
        rowsum[b * ND + d] = s;
        atomicAdd(&chansum[d], s);
        atomicAdd(&chansumsq[d], s2);
    }
}

// ---------------------------------------------------------------------------
// 3) norm_mean M[b,d] = (mean_r(img) - mu[d]) * rsqrt(var[d]+eps)
__global__ void k_norm_mean(const float* __restrict__ rowsum, const float* __restrict__ chansum,
                            const float* __restrict__ chansumsq, float* __restrict__ Mmat) {
    const int idx = blockIdx.x * blockDim.x + threadIdx.x; // 0..131071
    const int d = idx & (ND - 1);
    const float invN = 1.0f / NSAMP;
    const float mu  = chansum[d] * invN;
    const float var = chansumsq[d] * invN - mu * mu;
    const float rs  = rsqrtf(var + 1e-5f);
    Mmat[idx] = (rowsum[idx] * (1.0f / (float)NR) - mu) * rs;
}

// ---------------------------------------------------------------------------
// 4) generic WMMA GEMM (NN): C(M,N) = A(M,K) @ B(K,N) + bias[N], optional ReLU.
//    One 16x16 output tile per wave; K-loop in steps of 4 using v_wmma_f32_16x16x4_f32.
//    A operand: lanes 0-15 hold (K=0,1), lanes 16-31 hold (K=2,3) -> one b64 load/lane.
__global__ void k_gemm_nn(const float* __restrict__ A, const float* __restrict__ B,
                          const float* __restrict__ bias, float* __restrict__ C,
                          int M, int N, int K, int do_relu) {
    const int lane = threadIdx.x & 31;
    const int half = lane >> 4;
    const int l16  = lane & 15;
    const int wavesPerBlock = blockDim.x >> 5;
    const int wave   = blockIdx.x * wavesPerBlock + (threadIdx.x >> 5);
    const int nwaves = gridDim.x * wavesPerBlock;
    const int ntN    = N >> 4;
    const int ntiles = (M >> 4) * ntN;
    for (int t = wave; t < ntiles; t += nwaves) {
        const int mb = (t / ntN) << 4;
        const int nb = (t % ntN) << 4;
        v8f acc = {};
        const float* arow = A + (size_t)(mb + l16) * K;
        const float* bcol = B + nb + l16;
        for (int k = 0; k < K; k += 4) {
            const int kk = k + 2 * half;
            v2f a = *(const v2f*)(arow + kk);
            v2f b;
            b.x = bcol[(size_t)kk * N];
            b.y = bcol[(size_t)(kk + 1) * N];
            acc = __builtin_amdgcn_wmma_f32_16x16x4_f32(
                false, a, false, b, (short)0, acc, false, false);
        }
        const float bn = bias[nb + l16];
        float* ccol = C + nb + l16;
        #pragma unroll
        for (int v = 0; v < 8; ++v) {
            float val = acc[v] + bn;
            if (do_relu) val = fmaxf(val, 0.0f);
            ccol[(size_t)(mb + v + 8 * half) * N] = val;
        }
    }
}

// ---------------------------------------------------------------------------
// 5) build per-caption factor arrays:
//    capn = cap_repr * rnorm ; A=(1+g)*capn ; P=(1+g)^2 ; Q=(1+g)*beta
//    r[c]=sum beta^2 ; sbeta[c]=sum beta*capn
__global__ void k_prepare(const float* __restrict__ cap_repr, const float* __restrict__ rnorm,
                          const float* __restrict__ gamma, const float* __restrict__ beta,
                          float* __restrict__ Aarr, float* __restrict__ Parr,
                          float* __restrict__ Qarr, float* __restrict__ rvec,
                          float* __restrict__ sbeta) {
    __shared__ float redr[256];
    __shared__ float reds[256];
    const int c   = blockIdx.x;
    const int tid = threadIdx.x;
    const float rn = rnorm[c];
    float rr = 0.0f, sb = 0.0f;
    for (int j = 0; j < 4; ++j) {
        const int idx = c * ND + j * 256 + tid;
        const float g1 = 1.0f + gamma[idx];
        const float bb = beta[idx];
        const float cn = cap_repr[idx] * rn;
        Aarr[idx] = g1 * cn;
        Parr[idx] = g1 * g1;
        Qarr[idx] = g1 * bb;
        rr += bb * bb;
        sb += bb * cn;
    }
    redr[tid] = rr; reds[tid] = sb;
    __syncthreads();
    for (int s = 128; s > 0; s >>= 1) {
        if (tid < s) { redr[tid] += redr[tid + s]; reds[tid] += reds[tid + s]; }
        __syncthreads();
    }
    if (tid == 0) { rvec[c] = redr[0]; sbeta[c] = reds[0]; }
}

// ---------------------------------------------------------------------------
// 6) fused triple GEMM (NT) + rsqrt epilogue:
//    G1 = M @ A^T, G2 = M^2 @ P^T, G3 = M @ Q^T  (all 128x128, K=1024)
//    sims[i,c] = (G1 + sbeta[c]) * rsqrt(G2 + 2*G3 + r[c])
__global__ void k_sims(const float* __restrict__ Mmat, const float* __restrict__ Aarr,
                       const float* __restrict__ Parr, const float* __restrict__ Qarr,
                       const float* __restrict__ rvec, const float* __restrict__ sbeta,
                       float* __restrict__ out) {
    const int lane = threadIdx.x & 31;
    const int half = lane >> 4;
    const int l16  = lane & 15;
    const int wave = blockIdx.x * (blockDim.x >> 5) + (threadIdx.x >> 5); // 0..63
    const int ib = (wave >> 3) << 4;   // image-row tile base
    const int cb = (wave & 7) << 4;    // caption-col tile base
    v8f acc1 = {}, acc2 = {}, acc3 = {};
    const float* mrow = Mmat + (size_t)(ib + l16) * ND;
    const float* arow = Aarr + (size_t)(cb + l16) * ND;
    const float* prow = Parr + (size_t)(cb + l16) * ND;
    const float* qrow = Qarr + (size_t)(cb + l16) * ND;
    for (int k = 0; k < ND; k += 4) {
        const int kk = k + 2 * half;
        v2f am  = *(const v2f*)(mrow + kk);
        v2f am2 = am * am;
        v2f ba  = *(const v2f*)(arow + kk);
        v2f bp  = *(const v2f*)(prow + kk);
        v2f bq  = *(const v2f*)(qrow + kk);
        acc1 = __builtin_amdgcn_wmma_f32_16x16x4_f32(false, am,  false, ba, (short)0, acc1, false, false);
        acc2 = __builtin_amdgcn_wmma_f32_16x16x4_f32(false, am2, false, bp, (short)0, acc2, false, false);
        acc3 = __builtin_amdgcn_wmma_f32_16x16x4_f32(false, am,  false, bq, (short)0, acc3, false, false);
    }
    const int c = cb + l16;
    const float sb = sbeta[c];
    const float rc = rvec[c];
    #pragma unroll
    for (int v = 0; v < 8; ++v) {
        const int i = ib + v + 8 * half;
        const float num = acc1[v] + sb;
        const float den = acc2[v] + 2.0f * acc3[v] + rc;
        out[i * NB + c] = num * rsqrtf(den);
    }
}

// ---------------------------------------------------------------------------
extern "C" void kernel_launch(void* const* d_in, const int* in_sizes, int n_in,
                              void* d_out, int out_size, void* d_ws, size_t ws_size,
                              hipStream_t stream) {
    const float* img_embed = (const float*)d_in[0];
    const float* cap_embed = (const float*)d_in[1];
    const int*   lens      = (const int*)  d_in[2];
    const float* Wg1 = (const float*)d_in[3];
    const float* bg1 = (const float*)d_in[4];
    const float* Wg2 = (const float*)d_in[5];
    const float* bg2 = (const float*)d_in[6];
    const float* Wb1 = (const float*)d_in[7];
    const float* bb1 = (const float*)d_in[8];
    const float* Wb2 = (const float*)d_in[9];
    const float* bb2 = (const float*)d_in[10];
    float* out = (float*)d_out;

    // workspace layout (floats)
    float* ws = (float*)d_ws;
    float* cap_repr  = ws;                    // 131072
    float* rnorm     = cap_repr  + NB * ND;   // 128
    float* rowsum    = rnorm     + NB;        // 131072
    float* chansum   = rowsum    + NB * ND;   // 1024
    float* chansumsq = chansum   + ND;        // 1024
    float* Mmat      = chansumsq + ND;        // 131072
    float* h_g       = Mmat      + NB * ND;   // 16384
    float* h_b       = h_g       + NB * NH;   // 16384
    float* gamma_    = h_b       + NB * NH;   // 131072
    float* beta_     = gamma_    + NB * ND;   // 131072
    float* Aarr      = beta_     + NB * ND;   // 131072
    float* Parr      = Aarr      + NB * ND;   // 131072
    float* Qarr      = Parr      + NB * ND;   // 131072
    float* rvec      = Qarr      + NB * ND;   // 128
    float* sbeta     = rvec      + NB;        // 128

    // 0) zero atomic accumulators (ws not re-poisoned between replays)
    k_zero<<<8, 256, 0, stream>>>(chansum, 2 * ND);

    // 1) caption pooling + per-caption inverse norm
    k_cap_pool<<<NB, 256, 0, stream>>>(cap_embed, lens, cap_repr, rnorm);

    // 2) img region sums + BN channel stats (single pass over img_embed)
    k_img_stats<<<NB, 256, 0, stream>>>(img_embed, rowsum, chansum, chansumsq);

    // 3) norm_mean
    k_norm_mean<<<(NB * ND) / 256, 256, 0, stream>>>(rowsum, chansum, chansumsq, Mmat);

    // 4) MLPs: gamma = relu(cap@Wg1+bg1)@Wg2+bg2 ; beta likewise
    //    layer1: M=128,N=128,K=1024 -> 64 tiles -> 8 blocks x 8 waves
    k_gemm_nn<<<8, 256, 0, stream>>>(cap_repr, Wg1, bg1, h_g, NB, NH, ND, 1);
    k_gemm_nn<<<8, 256, 0, stream>>>(cap_repr, Wb1, bb1, h_b, NB, NH, ND, 1);
    //    layer2: M=128,N=1024,K=128 -> 512 tiles -> 64 blocks x 8 waves
    k_gemm_nn<<<64, 256, 0, stream>>>(h_g, Wg2, bg2, gamma_, NB, ND, NH, 0);
    k_gemm_nn<<<64, 256, 0, stream>>>(h_b, Wb2, bb2, beta_, NB, ND, NH, 0);

    // 5) factor arrays + per-caption scalars
    k_prepare<<<NB, 256, 0, stream>>>(cap_repr, rnorm, gamma_, beta_,
                                      Aarr, Parr, Qarr, rvec, sbeta);

    // 6) fused triple-GEMM similarity: 64 tiles -> 8 blocks x 8 waves
    k_sims<<<8, 256, 0, stream>>>(Mmat, Aarr, Parr, Qarr, rvec, sbeta, out);

    (void)in_sizes; (void)n_in; (void)out_size; (void)ws_size;
}